// RankingLoss_62130996904004
// MI455X (gfx1250) — compile-verified
//
#include <hip/hip_runtime.h>

// Spearman ranking loss for (256, 32768) fp32 rows on gfx1250 (MI455X).
//
// One workgroup (1024 thr = 32 wave32) per row, saturating a WGP. The TDM
// (tensor_load_to_lds + s_wait_tensorcnt) DMAs each 128KB row straight into
// LDS; the row is expanded in place to 64-bit (orderedKey<<32 | idx) entries
// (256KB) and bitonic-sorted entirely inside the 320KB LDS. Prediction ranks
// park in a 64KB ushort table; after sorting targets, S = sum (pr-tr)^2 is
// accumulated exactly in u64 and the closed-form Spearman identity
//   corr = (c - S/2) / (c + 1e-8),  c = n(n^2-1)/12
// gives the loss. The target row is prefetched into the 192MB L2 during the
// prediction-phase sort so the second TDM streams from L2, not HBM.

#define ROWS 256
#define N    32768
#define NT   1024
#define PER  (N / NT)                       // 32 elements per thread
#define LDS_BYTES ((size_t)N * 8 + (size_t)N * 2)   // 256KB keys + 64KB ranks = 320KB

typedef unsigned int u32x4 __attribute__((ext_vector_type(4)));
typedef int          i32x4 __attribute__((ext_vector_type(4)));
typedef int          i32x8 __attribute__((ext_vector_type(8)));

// ---- Tensor Data Mover: DMA one row of 32768 f32 from global into LDS ----
__device__ __forceinline__ void tdm_load_row_f32(const float* gsrc,
                                                 unsigned lds_byte_addr) {
  unsigned long long ga = (unsigned long long)(uintptr_t)gsrc;

  // D# group 0: count=1 (valid user descriptor), lds_addr, global_addr, type=2
  u32x4 g0;
  g0.x = 1u;
  g0.y = lds_byte_addr;
  g0.z = (unsigned)ga;                                       // global_addr[31:0]
  g0.w = (unsigned)((ga >> 32) & 0x01FFFFFFu) | (2u << 30);  // [56:32] | type=2

  // D# group 1: data_size=4B, tensor_dim0=32768, tensor_dim1=1,
  // tile_dim0=32768, tile_dim1=1, strides=32768.
  i32x8 g1;
  g1[0] = (int)(2u << 16);        // data_size = 2 -> 4 bytes; no multicast/flags
  g1[1] = (int)(0x8000u << 16);   // tensor_dim0[15:0] = 32768
  g1[2] = (int)(1u << 16);        // tensor_dim0 hi = 0, tensor_dim1 = 1
  g1[3] = (int)(0x8000u << 16);   // tile_dim0 = 32768
  g1[4] = 1;                      // tile_dim1 = 1
  g1[5] = (int)32768;             // tensor_dim0_stride lo32
  g1[6] = (int)(0x8000u << 16);   // stride0 hi=0, tensor_dim1_stride lo16
  g1[7] = 0;

  i32x4 z4 = {0, 0, 0, 0};
#if defined(__clang_major__) && (__clang_major__ >= 23)
  i32x8 z8 = {0, 0, 0, 0, 0, 0, 0, 0};
  __builtin_amdgcn_tensor_load_to_lds(g0, g1, z4, z4, z8, 0);
#else
  __builtin_amdgcn_tensor_load_to_lds(g0, g1, z4, z4, 0);
#endif
  __builtin_amdgcn_s_wait_tensorcnt(0);
}

__global__ void __launch_bounds__(NT)
spearman_row_kernel(const float* __restrict__ pred,
                    const float* __restrict__ targ,
                    unsigned long long* __restrict__ rowS) {
  extern __shared__ unsigned char smem[];
  unsigned long long* keys   = (unsigned long long*)smem;              // [0, 256K)
  float*              fstage = (float*)(smem + (size_t)N * 4);         // [128K, 256K)
  unsigned short*     pranks = (unsigned short*)(smem + (size_t)N * 8);// [256K, 320K)

  const int row = blockIdx.x;
  const int tid = threadIdx.x;
  const unsigned lds_base   = (unsigned)(unsigned long long)(uintptr_t)smem;
  const unsigned fstage_lds = lds_base + (unsigned)(N * 4);

  for (int phase = 0; phase < 2; ++phase) {
    const float* src = (phase == 0) ? (pred + (size_t)row * N)
                                    : (targ + (size_t)row * N);

    // --- async DMA row into LDS staging (one wave issues; TDM ignores EXEC) ---
    if (tid == 0) {
      tdm_load_row_f32(src, fstage_lds);
    }
    if (phase == 0) {
      // Warm the 192MB L2 with the target row while we sort predictions:
      // 1024 threads x 1 cacheline (32 f32 = 128B) = the whole 128KB row.
      // global_prefetch_b8: no LOADcnt, no data return, pure L2 warm-up.
      __builtin_prefetch(targ + (size_t)row * N + (size_t)tid * 32, 0, 1);
    }
    __syncthreads();

    // --- in-place expand f32 -> packed 64-bit sort keys via registers ---
    float v[PER];
#pragma unroll
    for (int u = 0; u < PER; ++u) v[u] = fstage[tid + u * NT];
    __syncthreads();
#pragma unroll
    for (int u = 0; u < PER; ++u) {
      int i = tid + u * NT;
      unsigned b = __float_as_uint(v[u]);
      b = (b & 0x80000000u) ? ~b : (b | 0x80000000u);   // order-preserving map
      keys[i] = ((unsigned long long)b << 32) | (unsigned)i;
    }

    // --- bitonic sort ascending, fully LDS-resident ---
    for (unsigned k = 2; k <= (unsigned)N; k <<= 1) {
      for (unsigned j = k >> 1; j > 0; j >>= 1) {
        __syncthreads();
        for (unsigned t = tid; t < (unsigned)(N / 2); t += NT) {
          unsigned i   = ((t & ~(j - 1u)) << 1) | (t & (j - 1u));
          unsigned ixj = i | j;
          unsigned long long a = keys[i];
          unsigned long long c = keys[ixj];
          bool up = ((i & k) == 0u);
          if ((a > c) == up) { keys[i] = c; keys[ixj] = a; }
        }
      }
    }
    __syncthreads();

    if (phase == 0) {
      // park prediction ranks: pranks[original_idx] = sorted position
      for (int p = tid; p < N; p += NT)
        pranks[(unsigned)keys[p] & 0x7FFFu] = (unsigned short)p;
      __syncthreads();
    } else {
      // S = sum over elements of (pr - tr)^2, exact in u64
      unsigned long long acc = 0;
      for (int p = tid; p < N; p += NT) {
        int idx = (int)((unsigned)keys[p] & 0x7FFFu);
        int d   = p - (int)pranks[idx];
        acc += (unsigned long long)((long long)d * (long long)d);
      }
      __syncthreads();
      keys[tid] = acc;                       // reuse keys LDS as u64 scratch
      for (int s = NT / 2; s > 0; s >>= 1) {
        __syncthreads();
        if (tid < s) keys[tid] += keys[tid + s];
      }
      if (tid == 0) rowS[row] = keys[0];
    }
  }
}

__global__ void finalize_kernel(const unsigned long long* __restrict__ rowS,
                                float* __restrict__ out) {
  __shared__ double sh[ROWS];
  const int t = threadIdx.x;
  const double nD = (double)N;
  const double c  = nD * (nD * nD - 1.0) / 12.0;     // sum of centered-rank squares
  const double S  = (double)rowS[t];
  const double corr = (c - 0.5 * S) / (c + 1e-8);
  sh[t] = 1.0 - corr;
  __syncthreads();
  for (int s = ROWS / 2; s > 0; s >>= 1) {
    if (t < s) sh[t] += sh[t + s];
    __syncthreads();
  }
  if (t == 0) out[0] = (float)(sh[0] / (double)ROWS);
}

extern "C" void kernel_launch(void* const* d_in, const int* in_sizes, int n_in,
                              void* d_out, int out_size, void* d_ws, size_t ws_size,
                              hipStream_t stream) {
  const float* pred = (const float*)d_in[0];
  const float* targ = (const float*)d_in[1];
  unsigned long long* rowS = (unsigned long long*)d_ws;   // 256 * 8B = 2KB scratch
  float* out = (float*)d_out;

  spearman_row_kernel<<<dim3(ROWS), dim3(NT), (unsigned)LDS_BYTES, stream>>>(
      pred, targ, rowS);
  finalize_kernel<<<dim3(1), dim3(ROWS), 0, stream>>>(rowS, out);
}